// FastAttention_26671746908377
// MI455X (gfx1250) — compile-verified
//
#include <hip/hip_runtime.h>
#include <cstdint>

typedef float v2f __attribute__((ext_vector_type(2)));
typedef float v8f __attribute__((ext_vector_type(8)));
typedef uint32_t u32;

#define B_SZ 4
#define H_SZ 8
#define T_SZ 8192
#define BH_SZ (B_SZ * H_SZ)          // 32
#define NCHUNK 16
#define TCHUNK (T_SZ / NCHUNK)       // 512
#define WAVES 2
#define BLOCK (WAVES * 32)           // 64
#define NT (TCHUNK / (16 * WAVES))   // 16 tiles per wave
#define LDP 66                       // padded LDS row stride in floats
#define TILE_F (16 * LDP)            // floats per 16-row staged tile

// ---------------------------------------------------------------------------
// fp32 WMMA 16x16x4: D = A(16x4) * B(4x16) + C
// A frag (v2f): lane = r + 16*h holds A[r][k0+2h], A[r][k0+2h+1]
// B frag (v2f): lane = n + 16*h holds B[k0+2h][n], B[k0+2h+1][n]
// C/D frag (v8f): element i on lane = n + 16*h is C[i + 8h][n]
// ---------------------------------------------------------------------------
__device__ __forceinline__ v8f wmma4(v2f a, v2f b, v8f c) {
  return __builtin_amdgcn_wmma_f32_16x16x4_f32(
      false, a, false, b, (short)0, c, false, false);
}

// Generic pointers to LDS: low 32 bits are the LDS byte offset (ISA 10.2,
// aperture mapping truncates to addr[31:0]).
__device__ __forceinline__ u32 lds_addr_of(const void* p) {
  return (u32)(uintptr_t)p;
}

// Async-copy one 16x64 fp32 tile (row-major, 256B rows) from global memory
// into LDS with a 66-float (264B) padded row stride.  8 x b128: each
// instruction moves 2 rows (32 lanes x 16B), fully coalesced, no VGPR data.
// Tracked by ASYNCcnt; caller waits with s_wait_asynccnt.
__device__ __forceinline__ void async_tile_16x64(u32 lds0, const char* g0) {
#pragma unroll
  for (int i = 0; i < 8; ++i) {
    u32 l = lds0 + (u32)(i * (2 * LDP * 4));   // 2 rows * 264B
    const char* g = g0 + i * 512;              // 2 rows * 256B
    asm volatile("global_load_async_to_lds_b128 %0, %1, off"
                 :: "v"(l), "v"(g) : "memory");
  }
}

__device__ __forceinline__ void wait_async_all() {
  asm volatile("s_wait_asynccnt 0x0" ::: "memory");
}

// Per-row LayerNorm (+ optional L2 normalize) on a 16x64 tile held in
// A-fragment layout. Row reduction = in-lane sum + shfl_xor(16).
// Consumes every staged LDS value -> safe point to overwrite the tile buffer.
__device__ __forceinline__ void ln_rows(v2f (&x)[16], const float* gbL,
                                        bool do_l2, int h) {
  float s = 0.f, s2 = 0.f;
#pragma unroll
  for (int j = 0; j < 16; ++j) {
    s  += x[j].x + x[j].y;
    s2 += x[j].x * x[j].x + x[j].y * x[j].y;
  }
  s  += __shfl_xor(s, 16);
  s2 += __shfl_xor(s2, 16);
  const float mu  = s * (1.0f / 64.0f);
  const float var = s2 * (1.0f / 64.0f) - mu * mu;
  const float rs  = rsqrtf(var + 1e-5f);
  float n2 = 0.f;
#pragma unroll
  for (int j = 0; j < 16; ++j) {
    v2f g = *(const v2f*)(gbL + 4 * j + 2 * h);
    v2f b = *(const v2f*)(gbL + 64 + 4 * j + 2 * h);
    x[j].x = (x[j].x - mu) * rs * g.x + b.x;
    x[j].y = (x[j].y - mu) * rs * g.y + b.y;
    n2 += x[j].x * x[j].x + x[j].y * x[j].y;
  }
  if (do_l2) {
    n2 += __shfl_xor(n2, 16);
    const float inv = 1.0f / fmaxf(sqrtf(n2), 1e-12f);
#pragma unroll
    for (int j = 0; j < 16; ++j) { x[j].x *= inv; x[j].y *= inv; }
  }
}

// out(16x64) = 0.1 * exp(clip(A(16x64) @ proj(64x64), -15, 15)); C-fragments.
__device__ __forceinline__ void feat_map(const v2f (&a)[16], const float* projL,
                                         int lane, v8f (&out)[4]) {
  const int h = lane >> 4, n = lane & 15;
#pragma unroll
  for (int nt = 0; nt < 4; ++nt) {
    v8f acc = {};
#pragma unroll
    for (int j = 0; j < 16; ++j) {
      v2f b;
      b.x = projL[(4 * j + 2 * h)     * LDP + nt * 16 + n];
      b.y = projL[(4 * j + 2 * h + 1) * LDP + nt * 16 + n];
      acc = wmma4(a[j], b, acc);
    }
#pragma unroll
    for (int i = 0; i < 8; ++i) {
      float t = fminf(fmaxf(acc[i], -15.0f), 15.0f);
      acc[i] = 0.1f * __expf(t);
    }
    out[nt] = acc;
  }
}

// ---------------------------------------------------------------------------
// Pass 1: kv_partial(bh,chunk) = 0.1 * sum_t kp[t]^T vn[t]
// ---------------------------------------------------------------------------
__global__ __launch_bounds__(BLOCK) void kv_pass(
    const float* __restrict__ k, const float* __restrict__ v,
    const float* __restrict__ proj, const float* __restrict__ gamma,
    const float* __restrict__ beta, float* __restrict__ kv_part) {
  __shared__ float projL[64 * LDP];
  __shared__ float gbL[128];
  __shared__ float tiles[WAVES][4][TILE_F];  // [wave][kbuf,vbuf,kpL,vnL]

  const int bh = blockIdx.x, chunk = blockIdx.y;
  const int tid = threadIdx.x, wave = tid >> 5, lane = tid & 31;
  const int h = lane >> 4, r = lane & 15;
  const u32 lsub = (u32)(h * (LDP * 4) + r * 16);  // per-lane LDS sub-offset
  const int gsub = h * 256 + r * 16;               // per-lane global sub-offset

  for (int i = tid; i < 4096; i += BLOCK)
    projL[(i >> 6) * LDP + (i & 63)] = proj[i];
  if (tid < 64) { gbL[tid] = gamma[tid]; gbL[64 + tid] = beta[tid]; }
  __syncthreads();

  float* kb  = tiles[wave][0];
  float* vb  = tiles[wave][1];
  float* kpL = tiles[wave][2];
  float* vnL = tiles[wave][3];
  const u32 kb_l = lds_addr_of(kb) + lsub;
  const u32 vb_l = lds_addr_of(vb) + lsub;

  const size_t base = (size_t)bh * T_SZ;
  const int tstart = chunk * TCHUNK + wave * 16;

  // prologue: prefetch tile 0
  async_tile_16x64(kb_l, (const char*)(k + (base + tstart) * 64) + gsub);
  async_tile_16x64(vb_l, (const char*)(v + (base + tstart) * 64) + gsub);

  v8f kvacc[4][4];
#pragma unroll
  for (int mt = 0; mt < 4; ++mt)
#pragma unroll
    for (int nt = 0; nt < 4; ++nt) kvacc[mt][nt] = {};

  for (int it = 0; it < NT; ++it) {
    const int t0 = tstart + it * (WAVES * 16);
    wait_async_all();

    v2f kx[16];
#pragma unroll
    for (int j = 0; j < 16; ++j) kx[j] = *(const v2f*)(kb + r * LDP + 4 * j + 2 * h);
    ln_rows(kx, gbL, true, h);          // consumes all kbuf reads
    v2f vx[16];
#pragma unroll
    for (int j = 0; j < 16; ++j) vx[j] = *(const v2f*)(vb + r * LDP + 4 * j + 2 * h);
    ln_rows(vx, gbL, false, h);         // consumes all vbuf reads

    if (it + 1 < NT) {                  // prefetch next tile behind the compute
      const int t1 = t0 + WAVES * 16;
      async_tile_16x64(kb_l, (const char*)(k + (base + t1) * 64) + gsub);
      async_tile_16x64(vb_l, (const char*)(v + (base + t1) * 64) + gsub);
    }

    v8f kp[4];
    feat_map(kx, projL, lane, kp);
#pragma unroll
    for (int nt = 0; nt < 4; ++nt)
#pragma unroll
      for (int i = 0; i < 8; ++i)
        kpL[(i + 8 * h) * LDP + nt * 16 + r] = kp[nt][i];
#pragma unroll
    for (int j = 0; j < 16; ++j)
      *(v2f*)(vnL + r * LDP + 4 * j + 2 * h) = vx[j];

    // kv += kp^T (64x16) @ vn (16x64)
#pragma unroll
    for (int mt = 0; mt < 4; ++mt) {
#pragma unroll
      for (int j = 0; j < 4; ++j) {
        v2f a;
        a.x = kpL[(4 * j + 2 * h)     * LDP + mt * 16 + r];
        a.y = kpL[(4 * j + 2 * h + 1) * LDP + mt * 16 + r];
#pragma unroll
        for (int nt = 0; nt < 4; ++nt) {
          v2f b;
          b.x = vnL[(4 * j + 2 * h)     * LDP + nt * 16 + r];
          b.y = vnL[(4 * j + 2 * h + 1) * LDP + nt * 16 + r];
          kvacc[mt][nt] = wmma4(a, b, kvacc[mt][nt]);
        }
      }
    }
  }

  // deterministic phased cross-wave reduction (tile staging is dead; reuse it)
  __syncthreads();
  float* kvred = &tiles[0][0][0];
  for (int w = 0; w < WAVES; ++w) {
    if (wave == w) {
#pragma unroll
      for (int mt = 0; mt < 4; ++mt)
#pragma unroll
        for (int nt = 0; nt < 4; ++nt)
#pragma unroll
          for (int i = 0; i < 8; ++i) {
            const int m = mt * 16 + i + 8 * h;
            const int n = nt * 16 + r;
            if (w == 0) kvred[m * 64 + n] = kvacc[mt][nt][i];
            else        kvred[m * 64 + n] += kvacc[mt][nt][i];
          }
    }
    __syncthreads();
  }
  float* dst = kv_part + ((size_t)bh * NCHUNK + chunk) * 4096;
  for (int i = tid; i < 4096; i += BLOCK) dst[i] = 0.1f * kvred[i];
}

// ---------------------------------------------------------------------------
// Pass 2: out = LN( (qp @ kv) * 0.1 / max(sum(qp*kp), 1e-6) )
// ---------------------------------------------------------------------------
__global__ __launch_bounds__(BLOCK) void out_pass(
    const float* __restrict__ q, const float* __restrict__ k,
    const float* __restrict__ proj, const float* __restrict__ gamma,
    const float* __restrict__ beta, const float* __restrict__ kv_part,
    float* __restrict__ out) {
  __shared__ float projL[64 * LDP];
  __shared__ float kvL[64 * LDP];
  __shared__ float gbL[128];
  __shared__ float tiles[WAVES][3][TILE_F];  // [wave][qbuf,kbuf,qpL]

  const int bh = blockIdx.x, chunk = blockIdx.y;
  const int tid = threadIdx.x, wave = tid >> 5, lane = tid & 31;
  const int h = lane >> 4, r = lane & 15;
  const u32 lsub = (u32)(h * (LDP * 4) + r * 16);
  const int gsub = h * 256 + r * 16;

  const float* part = kv_part + (size_t)bh * NCHUNK * 4096;
  for (int i = tid; i < 4096; i += BLOCK) {
    projL[(i >> 6) * LDP + (i & 63)] = proj[i];
    float acc = 0.f;
#pragma unroll
    for (int c = 0; c < NCHUNK; ++c) acc += part[c * 4096 + i];
    kvL[(i >> 6) * LDP + (i & 63)] = acc;
  }
  if (tid < 64) { gbL[tid] = gamma[tid]; gbL[64 + tid] = beta[tid]; }
  __syncthreads();

  float* qb  = tiles[wave][0];
  float* kb  = tiles[wave][1];
  float* qpL = tiles[wave][2];
  const u32 qb_l = lds_addr_of(qb) + lsub;
  const u32 kb_l = lds_addr_of(kb) + lsub;

  const size_t base = (size_t)bh * T_SZ;
  const int tstart = chunk * TCHUNK + wave * 16;

  async_tile_16x64(qb_l, (const char*)(q + (base + tstart) * 64) + gsub);
  async_tile_16x64(kb_l, (const char*)(k + (base + tstart) * 64) + gsub);

  for (int it = 0; it < NT; ++it) {
    const int t0 = tstart + it * (WAVES * 16);
    wait_async_all();

    v2f qx[16];
#pragma unroll
    for (int j = 0; j < 16; ++j) qx[j] = *(const v2f*)(qb + r * LDP + 4 * j + 2 * h);
    ln_rows(qx, gbL, true, h);
    v2f kx[16];
#pragma unroll
    for (int j = 0; j < 16; ++j) kx[j] = *(const v2f*)(kb + r * LDP + 4 * j + 2 * h);
    ln_rows(kx, gbL, true, h);

    if (it + 1 < NT) {
      const int t1 = t0 + WAVES * 16;
      async_tile_16x64(qb_l, (const char*)(q + (base + t1) * 64) + gsub);
      async_tile_16x64(kb_l, (const char*)(k + (base + t1) * 64) + gsub);
    }

    v8f qp[4], kp[4];
    feat_map(qx, projL, lane, qp);
    feat_map(kx, projL, lane, kp);

    // denom[t] = sum_m qp*kp ; C-frag rows t = i + 8h
    float dinv[8];
#pragma unroll
    for (int i = 0; i < 8; ++i) {
      float d = 0.f;
#pragma unroll
      for (int nt = 0; nt < 4; ++nt) d += qp[nt][i] * kp[nt][i];
      d += __shfl_xor(d, 1);
      d += __shfl_xor(d, 2);
      d += __shfl_xor(d, 4);
      d += __shfl_xor(d, 8);
      dinv[i] = 1.0f / fmaxf(d, 1e-6f);
    }

    // stage qp row-major, then qkv = qp(16x64) @ kv(64x64)
#pragma unroll
    for (int nt = 0; nt < 4; ++nt)
#pragma unroll
      for (int i = 0; i < 8; ++i)
        qpL[(i + 8 * h) * LDP + nt * 16 + r] = qp[nt][i];

    v8f acc[4];
#pragma unroll
    for (int nt = 0; nt < 4; ++nt) acc[nt] = {};
#pragma unroll
    for (int j = 0; j < 16; ++j) {
      v2f a = *(const v2f*)(qpL + r * LDP + 4 * j + 2 * h);
#pragma unroll
      for (int nt = 0; nt < 4; ++nt) {
        v2f b;
        b.x = kvL[(4 * j + 2 * h)     * LDP + nt * 16 + r];
        b.y = kvL[(4 * j + 2 * h + 1) * LDP + nt * 16 + r];
        acc[nt] = wmma4(a, b, acc[nt]);
      }
    }

    // scale, divide, fused final LayerNorm, store
#pragma unroll
    for (int i = 0; i < 8; ++i) {
      float sm = 0.f, sq = 0.f;
#pragma unroll
      for (int nt = 0; nt < 4; ++nt) {
        float val = acc[nt][i] * 0.1f * dinv[i];
        acc[nt][i] = val;
        sm += val;
        sq += val * val;
      }
      sm += __shfl_xor(sm, 1); sm += __shfl_xor(sm, 2);
      sm += __shfl_xor(sm, 4); sm += __shfl_xor(sm, 8);
      sq += __shfl_xor(sq, 1); sq += __shfl_xor(sq, 2);
      sq += __shfl_xor(sq, 4); sq += __shfl_xor(sq, 8);
      const float mu  = sm * (1.0f / 64.0f);
      const float var = sq * (1.0f / 64.0f) - mu * mu;
      const float rs  = rsqrtf(var + 1e-5f);
      float* orow = out + (base + t0 + i + 8 * h) * 64;
#pragma unroll
      for (int nt = 0; nt < 4; ++nt) {
        const int n = nt * 16 + r;
        orow[n] = (acc[nt][i] - mu) * rs * gbL[n] + gbL[64 + n];
      }
    }
  }
}

// ---------------------------------------------------------------------------
extern "C" void kernel_launch(void* const* d_in, const int* in_sizes, int n_in,
                              void* d_out, int out_size, void* d_ws, size_t ws_size,
                              hipStream_t stream) {
  (void)in_sizes; (void)n_in; (void)out_size; (void)ws_size;
  const float* q     = (const float*)d_in[0];
  const float* k     = (const float*)d_in[1];
  const float* v     = (const float*)d_in[2];
  const float* proj  = (const float*)d_in[3];
  const float* gamma = (const float*)d_in[4];
  const float* beta  = (const float*)d_in[5];
  float* out     = (float*)d_out;
  float* kv_part = (float*)d_ws;  // BH * NCHUNK * 64 * 64 floats = 8 MB

  dim3 grid(BH_SZ, NCHUNK);
  kv_pass<<<grid, BLOCK, 0, stream>>>(k, v, proj, gamma, beta, kv_part);
  out_pass<<<grid, BLOCK, 0, stream>>>(q, k, proj, gamma, beta, kv_part, out);
}